// AggrGATConv_38998303047881
// MI455X (gfx1250) — compile-verified
//
#include <hip/hip_runtime.h>
#include <hip/hip_bf16.h>

typedef __attribute__((ext_vector_type(2))) float v2f;
typedef __attribute__((ext_vector_type(8))) float v8f;

#define FEAT_PITCH 132  // 128 + 4: A-frag reads hit banks 4r+k -> conflict-free
#define W_PITCH    144  // 128 + 16: B-frag reads, half-waves 32 banks apart
#define NEG_SLOPE  0.2f

// ---------------------------------------------------------------------------
// Kernel 1: h = feat @ W via V_WMMA_F32_16X16X4_F32 (exact fp32).
// Block = 256 thr (8 waves) -> 64 output rows. W (64 KB) + feat tile (33 KB)
// staged in LDS (CDNA5 WGP has 320 KB). Each wave owns one 16-col tile and
// sweeps 4 row-tiles: 128 WMMA/wave, inner loop is LDS-only.
// ---------------------------------------------------------------------------
__global__ __launch_bounds__(256) void gat_gemm_wmma(
    const float* __restrict__ feat, const float* __restrict__ W,
    float* __restrict__ h, int N) {
  __shared__ float lfeat[64 * FEAT_PITCH];   // 33,792 B
  __shared__ float lW[128 * W_PITCH];        // 73,728 B

  const int rowbase = blockIdx.x * 64;
  const int tid = threadIdx.x;

  // Stage W: 4096 float4, coalesced; each wave iteration covers one 128-col row.
  for (int i = tid; i < 4096; i += 256) {
    const int row = i >> 5;                  // 0..127 (k)
    const int c4  = i & 31;                  // float4 col
    const float4 v = ((const float4*)(W + (size_t)row * 128))[c4];
    *(float4*)(lW + row * W_PITCH + c4 * 4) = v;
  }
  // Stage feat tile: 64 rows x 128 cols = 2048 float4, coalesced, clamp tail.
  for (int i = tid; i < 2048; i += 256) {
    const int r  = i >> 5;                   // 0..63
    const int c4 = i & 31;
    int rr = rowbase + r;
    if (rr >= N) rr = N - 1;
    const float4 v = ((const float4*)(feat + (size_t)rr * 128))[c4];
    *(float4*)(lfeat + r * FEAT_PITCH + c4 * 4) = v;
  }
  __syncthreads();

  const int wave = tid >> 5;                 // 0..7 -> col tile
  const int lane = tid & 31;
  const int r  = lane & 15;                  // A row / B,D col-in-tile
  const int kp = lane >> 4;                  // K-pair select
  const int col = wave * 16 + r;

  for (int rt = 0; rt < 4; ++rt) {           // 4 row tiles of 16
    const int arow = rt * 16 + r;
    v8f acc = {};
#pragma unroll
    for (int kk = 0; kk < 32; ++kk) {        // K = 128 = 32 * 4
      const int k0 = kk * 4 + 2 * kp;
      v2f a, b;
      a.x = lfeat[arow * FEAT_PITCH + k0];
      a.y = lfeat[arow * FEAT_PITCH + k0 + 1];
      b.x = lW[k0 * W_PITCH + col];
      b.y = lW[(k0 + 1) * W_PITCH + col];
      // (neg_a, A, neg_b, B, c_mod, C, reuse_a, reuse_b)
      acc = __builtin_amdgcn_wmma_f32_16x16x4_f32(
          false, a, false, b, (short)0, acc, false, false);
    }
    // D layout: lane holds col N=lane&15, VGPR v -> M = v + 8*(lane>>4)
    const int mbase = rowbase + rt * 16 + 8 * kp;
#pragma unroll
    for (int v = 0; v < 8; ++v) {
      const int row = mbase + v;
      if (row < N) h[(size_t)row * 128 + col] = acc[v];
    }
  }
}

// ---------------------------------------------------------------------------
// Kernel 2: el[n,h] = <h[n,h,:], attn_l[h,:]> ; er likewise
// ---------------------------------------------------------------------------
__global__ void gat_elr(const float* __restrict__ h,
                        const float* __restrict__ attn_l,
                        const float* __restrict__ attn_r,
                        float* __restrict__ el, float* __restrict__ er, int N) {
  const int idx = blockIdx.x * blockDim.x + threadIdx.x;   // (node, head)
  if (idx >= N * 4) return;
  const int node = idx >> 2, hh = idx & 3;
  const float4* hp = (const float4*)(h + (size_t)node * 128 + hh * 32);
  const float4* al = (const float4*)(attn_l + hh * 32);
  const float4* ar = (const float4*)(attn_r + hh * 32);
  float sl = 0.f, sr = 0.f;
#pragma unroll
  for (int i = 0; i < 8; ++i) {
    const float4 v = hp[i], a = al[i], b = ar[i];
    sl += v.x * a.x + v.y * a.y + v.z * a.z + v.w * a.w;
    sr += v.x * b.x + v.y * b.y + v.z * b.z + v.w * b.w;
  }
  el[idx] = sl;
  er[idx] = sr;
}

// ---------------------------------------------------------------------------
// Kernel 3: deterministic init: out = mean_h(bias), m = -inf, s = 0
// ---------------------------------------------------------------------------
__global__ void gat_init(const float* __restrict__ bias, float* __restrict__ out,
                         float* __restrict__ m, float* __restrict__ s, int N) {
  const int idx = blockIdx.x * blockDim.x + threadIdx.x;
  if (idx < N * 32) {
    const int d = idx & 31;
    out[idx] = 0.25f * (bias[d] + bias[32 + d] + bias[64 + d] + bias[96 + d]);
  }
  if (idx < N * 4) {
    m[idx] = -__builtin_inff();
    s[idx] = 0.f;
  }
}

// Sign-split float atomic max (correct for mixed signs, init = -inf).
__device__ __forceinline__ void atomicMaxF32(float* addr, float val) {
  if (val >= 0.f)
    atomicMax((int*)addr, __float_as_int(val));
  else
    atomicMin((unsigned int*)addr, __float_as_uint(val));
}

// ---------------------------------------------------------------------------
// Kernel 4: per-edge leaky-relu logits + segment max over dst
// ---------------------------------------------------------------------------
__global__ void gat_logit(const int* __restrict__ src, const int* __restrict__ dst,
                          const float* __restrict__ el, const float* __restrict__ er,
                          float* __restrict__ logits, float* __restrict__ m, int E) {
  const int idx = blockIdx.x * blockDim.x + threadIdx.x;   // (edge, head)
  if (idx >= E * 4) return;
  const int e = idx >> 2, hh = idx & 3;
  float v = el[src[e] * 4 + hh] + er[dst[e] * 4 + hh];
  v = (v > 0.f) ? v : NEG_SLOPE * v;
  logits[idx] = v;
  atomicMaxF32(&m[dst[e] * 4 + hh], v);
}

// ---------------------------------------------------------------------------
// Kernel 5: a = exp(e - m[dst]); segment sum s[dst]
// ---------------------------------------------------------------------------
__global__ void gat_exp(const int* __restrict__ dst, const float* __restrict__ m,
                        float* __restrict__ logits, float* __restrict__ s, int E) {
  const int idx = blockIdx.x * blockDim.x + threadIdx.x;
  if (idx >= E * 4) return;
  const int e = idx >> 2, hh = idx & 3;
  const float a = __expf(logits[idx] - m[dst[e] * 4 + hh]);
  logits[idx] = a;                                  // reuse buffer for a
  atomicAdd(&s[dst[e] * 4 + hh], a);
}

// ---------------------------------------------------------------------------
// Kernel 6: head-fused weighted scatter. One wave32 per edge, lane = dim d.
// out[dst,d] += (1/H) * sum_h h[src][h*32+d] * (a[e,h] / s[dst,h])
// ---------------------------------------------------------------------------
__global__ void gat_scatter(const int* __restrict__ src, const int* __restrict__ dst,
                            const float* __restrict__ h, const float* __restrict__ a,
                            const float* __restrict__ s, float* __restrict__ out, int E) {
  const int gid = blockIdx.x * blockDim.x + threadIdx.x;
  const int e = gid >> 5;
  const int d = gid & 31;
  if (e >= E) return;
  const int sn = src[e], dn = dst[e];
  const float* hs = h + (size_t)sn * 128;
  const float a0 = a[e * 4 + 0] / s[dn * 4 + 0];
  const float a1 = a[e * 4 + 1] / s[dn * 4 + 1];
  const float a2 = a[e * 4 + 2] / s[dn * 4 + 2];
  const float a3 = a[e * 4 + 3] / s[dn * 4 + 3];
  const float c = 0.25f * (hs[d] * a0 + hs[32 + d] * a1 +
                           hs[64 + d] * a2 + hs[96 + d] * a3);
  atomicAdd(&out[(size_t)dn * 32 + d], c);
}

// ---------------------------------------------------------------------------
extern "C" void kernel_launch(void* const* d_in, const int* in_sizes, int n_in,
                              void* d_out, int out_size, void* d_ws, size_t ws_size,
                              hipStream_t stream) {
  const float* feat   = (const float*)d_in[0];
  const float* W      = (const float*)d_in[1];
  const float* attn_l = (const float*)d_in[2];
  const float* attn_r = (const float*)d_in[3];
  const float* bias   = (const float*)d_in[4];
  const int*   src    = (const int*)d_in[5];
  const int*   dst    = (const int*)d_in[6];

  const int N = in_sizes[0] / 128;
  const int E = in_sizes[5];

  float* out = (float*)d_out;
  float* ws  = (float*)d_ws;
  float* h      = ws;                                // N*128
  float* el     = h  + (size_t)N * 128;              // N*4
  float* er     = el + (size_t)N * 4;                // N*4
  float* m      = er + (size_t)N * 4;                // N*4
  float* s      = m  + (size_t)N * 4;                // N*4
  float* logits = s  + (size_t)N * 4;                // E*4

  gat_gemm_wmma<<<(N + 63) / 64, 256, 0, stream>>>(feat, W, h, N);
  gat_elr<<<(N * 4 + 255) / 256, 256, 0, stream>>>(h, attn_l, attn_r, el, er, N);
  gat_init<<<(N * 32 + 255) / 256, 256, 0, stream>>>(bias, out, m, s, N);
  gat_logit<<<(E * 4 + 255) / 256, 256, 0, stream>>>(src, dst, el, er, logits, m, E);
  gat_exp<<<(E * 4 + 255) / 256, 256, 0, stream>>>(dst, m, logits, s, E);
  gat_scatter<<<((size_t)E * 32 + 255) / 256, 256, 0, stream>>>(src, dst, h, logits, s, out, E);
}